// AttentionPropagationLayer_25598005084608
// MI455X (gfx1250) — compile-verified
//
#include <hip/hip_runtime.h>
#include <hip/hip_bf16.h>

// CDNA5 / gfx1250. wave32. Uses V_WMMA_F32_16X16X4_F32 for the dot products.
typedef float v2f __attribute__((ext_vector_type(2)));
typedef float v8f __attribute__((ext_vector_type(8)));

#define WAVES_PER_BLOCK 8

__global__ __launch_bounds__(256) void
attn_prop_softmax_kernel(const float* __restrict__ x,   // (4, N, 64)
                         const float* __restrict__ W,   // (128,)
                         float* __restrict__ out,       // (N, 4)
                         int N) {
    // Per-wave staging: [wave][slot][row]; slot 0 = anchor, slot 1+k = x_k . w2
    __shared__ float smem[WAVES_PER_BLOCK][5][16];

    const int lane   = threadIdx.x & 31;
    const int waveId = threadIdx.x >> 5;
    const int col    = lane & 15;   // B-matrix column / A-matrix row index
    const int half   = lane >> 4;   // 0: K=0,1 ; 1: K=2,3 (within a 4-wide chunk)

    const int tile = blockIdx.x * WAVES_PER_BLOCK + waveId;
    const int n0   = tile * 16;
    if (n0 >= N) return;            // wave-uniform: EXEC stays all-1s for WMMA

    // ---- Build B registers once: 16 chunks of the 4x16 B matrix.
    // B[e,0] = w1[4c+e], B[e,1] = w2[4c+e], other columns zero.
    // Layout: VGPR0 holds rows K=0 (lanes 0-15) / K=2 (lanes 16-31), VGPR1 K=1/K=3.
    v2f bmat[16];
    const float* wsel = W + 64 * col;      // col==0 -> w1, col==1 -> w2
#pragma unroll
    for (int c = 0; c < 16; ++c) {
        v2f b; b.x = 0.0f; b.y = 0.0f;
        if (col < 2) {
            const float* p = wsel + 4 * c + 2 * half;
            b.x = p[0]; b.y = p[1];
        }
        bmat[c] = b;
    }

    // ---- A-tile base address for this lane (row = col, e-offset = 2*half).
    const float* xrow = x + ((size_t)(n0 + col)) * 64 + 2 * half;

#pragma unroll
    for (int k = 0; k < 4; ++k) {
        const float* xk = xrow + (size_t)k * (size_t)N * 64;

        // Stream the 16x64 f32 tile (non-temporal: 1 GB stream, read-once, > L2).
        v2f amat[16];
#pragma unroll
        for (int c = 0; c < 16; ++c)
            amat[c] = __builtin_nontemporal_load((const v2f*)(xk + 4 * c));

        // 16-deep accumulation chain over E=64 via v_wmma_f32_16x16x4_f32.
        v8f acc = {};
#pragma unroll
        for (int c = 0; c < 16; ++c)
            acc = __builtin_amdgcn_wmma_f32_16x16x4_f32(
                /*neg_a=*/false, amat[c], /*neg_b=*/false, bmat[c],
                /*c_mod=*/(short)0, acc, /*reuse_a=*/false, /*reuse_b=*/false);

        // D[row, col]: lane L<16 has rows 0..7 (VGPR j = row j), lane L+16 rows 8..15.
        if (col == 0 && k == 0) {          // column 0: anchor = x0 . w1
#pragma unroll
            for (int j = 0; j < 8; ++j)
                smem[waveId][0][j + 8 * half] = acc[j];
        }
        if (col == 1) {                    // column 1: x_k . w2
#pragma unroll
            for (int j = 0; j < 8; ++j)
                smem[waveId][1 + k][j + 8 * half] = acc[j];
        }
    }

    // Wave is lockstep; just drain LDS stores before cross-lane reads.
    asm volatile("s_wait_dscnt 0" ::: "memory");

    // ---- Finalize: lanes 0-15 each own one n; softmax over K=4, coalesced b128 store.
    if (lane < 16) {
        const int n = n0 + lane;
        if (n < N) {
            const float anchor = smem[waveId][0][lane];
            const float s0 = anchor + smem[waveId][1][lane];
            const float s1 = anchor + smem[waveId][2][lane];
            const float s2 = anchor + smem[waveId][3][lane];
            const float s3 = anchor + smem[waveId][4][lane];
            const float m  = fmaxf(fmaxf(s0, s1), fmaxf(s2, s3));
            const float e0 = __expf(s0 - m);
            const float e1 = __expf(s1 - m);
            const float e2 = __expf(s2 - m);
            const float e3 = __expf(s3 - m);
            const float inv = 1.0f / (e0 + e1 + e2 + e3);
            float4 r;
            r.x = e0 * inv; r.y = e1 * inv; r.z = e2 * inv; r.w = e3 * inv;
            *(float4*)(out + (size_t)n * 4) = r;
        }
    }
}

extern "C" void kernel_launch(void* const* d_in, const int* in_sizes, int n_in,
                              void* d_out, int out_size, void* d_ws, size_t ws_size,
                              hipStream_t stream) {
    const float* x = (const float*)d_in[0];   // (4, N, 64) f32
    const float* W = (const float*)d_in[1];   // (128,) f32
    float* out = (float*)d_out;               // (N, 4) f32

    const long long total = (long long)in_sizes[0];
    const int N = (int)(total / (4 * 64));    // 1,000,000

    const int tiles  = (N + 15) / 16;
    const int blocks = (tiles + WAVES_PER_BLOCK - 1) / WAVES_PER_BLOCK;

    attn_prop_softmax_kernel<<<blocks, 32 * WAVES_PER_BLOCK, 0, stream>>>(x, W, out, N);
}